// TestNet_50388556316741
// MI455X (gfx1250) — compile-verified
//
#include <hip/hip_runtime.h>

#define BLOCK 256
#define NITER 25
#define BIGF  1000000000000.0f

// CDNA5 async global->LDS path (ASYNCcnt-tracked), if toolchain exposes it.
#if defined(__HIP_DEVICE_COMPILE__) && defined(__gfx1250__)
#if __has_builtin(__builtin_amdgcn_global_load_async_to_lds_b64)
#define USE_ASYNC_LDS 1
#endif
#endif

#define AS1 __attribute__((address_space(1)))
#define AS3 __attribute__((address_space(3)))
typedef int v2i __attribute__((ext_vector_type(2)));

// Step-length ratio -v/dv: only bounds alpha (then scaled by 0.99), so a
// 1-ULP v_rcp_f32 is plenty and avoids the precise-div expansion
// (v_div_scale/v_div_fmas + s_denorm_mode toggles).
__device__ __forceinline__ float boundary_ratio(float v, float dv)
{
#if defined(__HIP_DEVICE_COMPILE__) && __has_builtin(__builtin_amdgcn_rcpf)
    return -v * __builtin_amdgcn_rcpf(dv);
#else
    return -v / dv;
#endif
}

__global__ __launch_bounds__(BLOCK, 1)
void lcp_ip_kernel(const float* __restrict__ pv, const float* __restrict__ mu,
                   float* __restrict__ out, int n)
{
#if USE_ASYNC_LDS
    __shared__ float lds_pv[2 * BLOCK];
#endif
    const int tid = threadIdx.x;
    const int gid = blockIdx.x * BLOCK + tid;
    int row = gid < n ? gid : (n - 1);   // n is a multiple of BLOCK; clamp anyway

    const float muv = mu[0];             // uniform -> s_load

    float pvx, pvy;
#if USE_ASYNC_LDS
    {
        // 8 bytes/lane, async DMA into this lane's private LDS slot.
        __builtin_amdgcn_global_load_async_to_lds_b64(
            (AS1 v2i*)(pv + 2 * row),
            (AS3 v2i*)(&lds_pv[2 * tid]),
            /*imm offset*/0, /*cpol*/0);
#if __has_builtin(__builtin_amdgcn_s_wait_asynccnt)
        __builtin_amdgcn_s_wait_asynccnt(0);
#else
        asm volatile("s_wait_asynccnt 0" ::: "memory");
#endif
        pvx = lds_pv[2 * tid + 0];
        pvy = lds_pv[2 * tid + 1];
    }
#else
    __builtin_prefetch(pv + 2 * row, 0, 0);   // global_prefetch_b8 on gfx1250
    pvx = pv[2 * row + 0];
    pvy = pv[2 * row + 1];
#endif

    // q = -prev + (-2, 1);  z = -q;  lam = s = 1
    const float q0 = -pvx - 2.0f;
    const float q1 = -pvy + 1.0f;
    float z0 = -q0, z1 = -q1;
    float l0 = 1.0f, l1 = 1.0f, l2 = 1.0f, l3 = 1.0f;
    float s0 = 1.0f, s1 = 1.0f, s2 = 1.0f, s3 = 1.0f;

#pragma unroll 1
    for (int it = 0; it < NITER; ++it) {
        // r_dual = z + q + lam@G ; G = [[0,-1],[1,0],[-1,0],[0,0]]
        const float rd0 = z0 + q0 + l1 - l2;
        const float rd1 = z1 + q1 - l0;
        // r_pri = z@G.T + s - lam@F.T ; F = [[0,0,0,0],[0,0,0,1],[0,0,0,1],[mu,-1,-1,0]]
        const float rp0 = -z1 + s0;
        const float rp1 =  z0 + s1 - l3;
        const float rp2 = -z0 + s2 - l3;
        const float rp3 =  s3 - muv * l0 + l1 + l2;
        // complementarity
        const float sig = 0.1f * 0.25f * (s0 * l0 + s1 * l1 + s2 * l2 + s3 * l3);
        const float rl0 = 1.0f / l0, rl1 = 1.0f / l1, rl2 = 1.0f / l2, rl3 = 1.0f / l3;
        const float D0 = s0 * rl0, D1 = s1 * rl1, D2 = s2 * rl2, D3 = s3 * rl3;
        // rhs = [-r_dual ; -r_pri + r_cent/lam] ; r_cent/lam = s - sig/lam
        const float rt0 = -rd0, rt1 = -rd1;
        const float rb0 = -rp0 + s0 - sig * rl0;
        const float rb1 = -rp1 + s1 - sig * rl1;
        const float rb2 = -rp2 + s2 - sig * rl2;
        const float rb3 = -rp3 + s3 - sig * rl3;

        // Schur complement:  (E - G G^T) dlam = rb - G rt,  dz = rt - G^T dlam
        // S row0 = (-1-D0,0,0,0)  -> dlam0 trivially; rows 1..3 -> 3x3 elimination.
        const float b0 = rb0 + rt1;
        const float b1 = rb1 - rt0;
        const float b2 = rb2 + rt0;
        const float dl0 = -b0 / (1.0f + D0);
        const float c3  = rb3 + muv * dl0;

        const float p1  = 1.0f + D1;
        const float p2  = 1.0f + D2;
        const float ip1 = 1.0f / p1;
        const float A22 = ip1 - p2;
        const float A23 = -1.0f - ip1;
        const float A32 = 1.0f + ip1;
        const float A33 = -D3 - ip1;
        const float c2p = b2 + b1 * ip1;
        const float c3p = c3 + b1 * ip1;
        const float iA22 = 1.0f / A22;
        const float f    = A32 * iA22;
        const float dl3  = (c3p - f * c2p) / (A33 - f * A23);
        const float dl2  = (c2p - A23 * dl3) * iA22;
        const float dl1  = (dl2 - dl3 - b1) * ip1;

        const float dz0 = rt0 - dl1 + dl2;
        const float dz1 = rt1 + dl0;

        // ds = -r_pri - dz@G.T + dlam@F.T
        const float ds0 = -rp0 + dz1;
        const float ds1 = -rp1 - dz0 + dl3;
        const float ds2 = -rp2 + dz0 + dl3;
        const float ds3 = -rp3 + muv * dl0 - dl1 - dl2;

        // fraction-to-boundary over (lam,dlam) and (s,ds)
        float t = BIGF;
        t = fminf(t, (dl0 < -1e-12f) ? boundary_ratio(l0, dl0) : BIGF);
        t = fminf(t, (dl1 < -1e-12f) ? boundary_ratio(l1, dl1) : BIGF);
        t = fminf(t, (dl2 < -1e-12f) ? boundary_ratio(l2, dl2) : BIGF);
        t = fminf(t, (dl3 < -1e-12f) ? boundary_ratio(l3, dl3) : BIGF);
        t = fminf(t, (ds0 < -1e-12f) ? boundary_ratio(s0, ds0) : BIGF);
        t = fminf(t, (ds1 < -1e-12f) ? boundary_ratio(s1, ds1) : BIGF);
        t = fminf(t, (ds2 < -1e-12f) ? boundary_ratio(s2, ds2) : BIGF);
        t = fminf(t, (ds3 < -1e-12f) ? boundary_ratio(s3, ds3) : BIGF);
        const float alpha = fminf(1.0f, 0.99f * t);

        z0 += alpha * dz0;  z1 += alpha * dz1;
        l0 += alpha * dl0;  l1 += alpha * dl1;  l2 += alpha * dl2;  l3 += alpha * dl3;
        s0 += alpha * ds0;  s1 += alpha * ds1;  s2 += alpha * ds2;  s3 += alpha * ds3;
    }

    if (gid < n) {
        float2 o; o.x = z0; o.y = z1;
        reinterpret_cast<float2*>(out)[gid] = o;   // coalesced b64 store
    }
}

extern "C" void kernel_launch(void* const* d_in, const int* in_sizes, int n_in,
                              void* d_out, int out_size, void* d_ws, size_t ws_size,
                              hipStream_t stream)
{
    const float* pv = (const float*)d_in[0];   // prev_vels, (B,2) f32
    const float* mu = (const float*)d_in[1];   // (1,) f32
    float* out = (float*)d_out;                // (B,2) f32
    const int n = in_sizes[0] / 2;
    const int grid = (n + BLOCK - 1) / BLOCK;
    lcp_ip_kernel<<<grid, BLOCK, 0, stream>>>(pv, mu, out, n);
}